// GATLayer_21921513079360
// MI455X (gfx1250) — compile-verified
//
#include <hip/hip_runtime.h>
#include <hip/hip_bf16.h>
#include <math.h>

typedef __attribute__((ext_vector_type(2))) float v2f;
typedef __attribute__((ext_vector_type(8))) float v8f;

#define N_NODES 50000
#define N_EDGES 800000
#define TOT_E   (N_EDGES + N_NODES)   // 850000 with self loops
#define IN_DIM  128
#define OUT_DIM 64
#define HEADS   8
#define HEAD_DIM 8
#define NEG_SLOPE 0.2f

// ---------------------------------------------------------------------------
// float atomic max via sign-aware int/uint atomics (correct for all floats,
// target initialized to -inf). Avoids CAS loops and float-atomicMax overload
// portability issues.
// ---------------------------------------------------------------------------
__device__ __forceinline__ void atomicMaxFloat(float* addr, float val) {
    if (val >= 0.0f) {
        atomicMax((int*)addr, __float_as_int(val));
    } else {
        atomicMin((unsigned int*)addr, __float_as_uint(val));
    }
}

// ---------------------------------------------------------------------------
// Kernel 0: init out=0, m=-inf, denom=0
// ---------------------------------------------------------------------------
__global__ __launch_bounds__(256) void init_kernel(float* __restrict__ out,
                                                   float* __restrict__ m,
                                                   float* __restrict__ denom) {
    int t = blockIdx.x * 256 + threadIdx.x;
    if (t < N_NODES * OUT_DIM) out[t] = 0.0f;
    if (t < N_NODES * HEADS) {
        m[t] = -INFINITY;
        denom[t] = 0.0f;
    }
}

// ---------------------------------------------------------------------------
// Kernel 1: h = x @ W^T  via V_WMMA_F32_16X16X4_F32.
// Block = 128 threads (4 waves). Block owns 16 rows of x (staged in LDS with
// row pitch 132 floats to spread A-fragment reads over 16 banks). Wave w
// computes the 16x16 tile at columns [16w, 16w+16). K loop: 32 WMMA ops.
//
// Fragment layouts (ISA 7.12.2, 32-bit):
//   A 16x4 : lane L, vgpr j -> M = L%16, K = 2*(L/16)+j
//   B 4x16 : lane L, vgpr j -> N = L%16, K = 2*(L/16)+j   (B[k][n] = W[n][k])
//   C/D    : lane L, vgpr v -> M = v + 8*(L/16), N = L%16
// ---------------------------------------------------------------------------
#define LDS_PITCH 132  // 16B-aligned pitch (528 B), bank-conflict free A reads

__global__ __launch_bounds__(128) void gemm_wmma_kernel(const float* __restrict__ x,
                                                        const float* __restrict__ W,
                                                        float* __restrict__ h) {
    __shared__ float tile[16 * LDS_PITCH];

    const int m_base = blockIdx.x * 16;
    const int tid    = threadIdx.x;

    // Cooperative load of the 16x128 fp32 x-tile (512 float4 loads).
    for (int i = tid; i < 16 * 32; i += 128) {
        int row = i >> 5;        // 32 float4 per row
        int c4  = i & 31;
        float4 v = ((const float4*)(x + (size_t)(m_base + row) * IN_DIM))[c4];
        *((float4*)&tile[row * LDS_PITCH + c4 * 4]) = v;
    }
    __syncthreads();

    const int wave   = tid >> 5;
    const int lane   = tid & 31;
    const int n_base = wave * 16;
    const int mrow   = lane & 15;        // M (for A) / N (for B) index
    const int kgrp   = (lane >> 4) * 2;  // K sub-offset: 0 or 2

    const float* wrow = W + (size_t)(n_base + mrow) * IN_DIM;

    v8f c = {};
#pragma unroll
    for (int k0 = 0; k0 < IN_DIM; k0 += 4) {
        v2f a = *(const v2f*)&tile[mrow * LDS_PITCH + k0 + kgrp];
        v2f b = *(const v2f*)&wrow[k0 + kgrp];
        c = __builtin_amdgcn_wmma_f32_16x16x4_f32(
                /*neg_a=*/false, a, /*neg_b=*/false, b,
                /*c_mod=*/(short)0, c, /*reuse_a=*/false, /*reuse_b=*/false);
    }

    const int mhalf = (lane >> 4) * 8;
#pragma unroll
    for (int v = 0; v < 8; ++v) {
        h[(size_t)(m_base + mhalf + v) * OUT_DIM + n_base + mrow] = c[v];
    }
}

// ---------------------------------------------------------------------------
// Kernel 2: per-(node,head) attention dots  s[n,h], d[n,h]
// ---------------------------------------------------------------------------
__global__ __launch_bounds__(256) void dots_kernel(const float* __restrict__ h,
                                                   const float* __restrict__ sa,
                                                   const float* __restrict__ da,
                                                   float* __restrict__ s,
                                                   float* __restrict__ d) {
    int t = blockIdx.x * 256 + threadIdx.x;
    if (t >= N_NODES * HEADS) return;
    int n  = t >> 3;
    int hh = t & 7;
    const float* hp  = h + (size_t)n * OUT_DIM + hh * HEAD_DIM;
    const float* sap = sa + hh * HEAD_DIM;
    const float* dap = da + hh * HEAD_DIM;
    float ss = 0.0f, dd = 0.0f;
#pragma unroll
    for (int k = 0; k < HEAD_DIM; ++k) {
        float v = hp[k];
        ss += v * sap[k];
        dd += v * dap[k];
    }
    s[t] = ss;
    d[t] = dd;
}

__device__ __forceinline__ void edge_row_col(int edge, const int* __restrict__ ei,
                                             int& row, int& col) {
    if (edge < N_EDGES) {
        row = ei[edge];
        col = ei[N_EDGES + edge];
    } else {
        row = col = edge - N_EDGES;   // self loop
    }
}

// ---------------------------------------------------------------------------
// Kernel 3: e = leaky_relu(s[row]+d[col]); segment atomic max into m[row]
// ---------------------------------------------------------------------------
__global__ __launch_bounds__(256) void edge_max_kernel(const int* __restrict__ ei,
                                                       const float* __restrict__ s,
                                                       const float* __restrict__ d,
                                                       float* __restrict__ m,
                                                       float* __restrict__ e) {
    int t = blockIdx.x * 256 + threadIdx.x;
    if (t >= TOT_E * HEADS) return;
    int edge = t >> 3;
    int hh   = t & 7;
    int row, col;
    edge_row_col(edge, ei, row, col);
    float v = s[row * HEADS + hh] + d[col * HEADS + hh];
    v = (v >= 0.0f) ? v : NEG_SLOPE * v;
    e[t] = v;
    atomicMaxFloat(&m[row * HEADS + hh], v);
}

// ---------------------------------------------------------------------------
// Kernel 4: ex = exp(e - m[row]); segment atomic sum into denom[row]
// ---------------------------------------------------------------------------
__global__ __launch_bounds__(256) void edge_sum_kernel(const int* __restrict__ ei,
                                                       const float* __restrict__ m,
                                                       float* __restrict__ denom,
                                                       float* __restrict__ e) {
    int t = blockIdx.x * 256 + threadIdx.x;
    if (t >= TOT_E * HEADS) return;
    int edge = t >> 3;
    int hh   = t & 7;
    int row, col;
    edge_row_col(edge, ei, row, col);
    float ex = __expf(e[t] - m[row * HEADS + hh]);
    e[t] = ex;
    atomicAdd(&denom[row * HEADS + hh], ex);
}

// ---------------------------------------------------------------------------
// Kernel 5: out[row] += (ex/denom[row]) * h[col]   (thread per edge-element;
// a wave covers 32 consecutive dims of one edge -> 128B coalesced gather and
// 128B-contiguous atomic burst)
// ---------------------------------------------------------------------------
__global__ __launch_bounds__(256) void edge_agg_kernel(const int* __restrict__ ei,
                                                       const float* __restrict__ e,
                                                       const float* __restrict__ denom,
                                                       const float* __restrict__ h,
                                                       float* __restrict__ out) {
    int t = blockIdx.x * 256 + threadIdx.x;
    if (t >= TOT_E * OUT_DIM) return;   // 54.4M < 2^31
    int edge = t >> 6;
    int idx  = t & 63;
    int hh   = idx >> 3;
    int row, col;
    edge_row_col(edge, ei, row, col);
    float w = e[edge * HEADS + hh] / denom[row * HEADS + hh];
    atomicAdd(&out[(size_t)row * OUT_DIM + idx],
              w * h[(size_t)col * OUT_DIM + idx]);
}

// ---------------------------------------------------------------------------
// Launch
// ---------------------------------------------------------------------------
extern "C" void kernel_launch(void* const* d_in, const int* in_sizes, int n_in,
                              void* d_out, int out_size, void* d_ws, size_t ws_size,
                              hipStream_t stream) {
    const float* x  = (const float*)d_in[0];
    const int*   ei = (const int*)d_in[1];
    const float* W  = (const float*)d_in[2];
    const float* sa = (const float*)d_in[3];
    const float* da = (const float*)d_in[4];
    float* out = (float*)d_out;

    // Workspace layout (floats): h | s | d | m | denom | e   (~46.4 MB)
    float* ws    = (float*)d_ws;
    float* h     = ws;
    float* s     = h + (size_t)N_NODES * OUT_DIM;
    float* dvec  = s + (size_t)N_NODES * HEADS;
    float* m     = dvec + (size_t)N_NODES * HEADS;
    float* denom = m + (size_t)N_NODES * HEADS;
    float* e     = denom + (size_t)N_NODES * HEADS;

    init_kernel<<<(N_NODES * OUT_DIM + 255) / 256, 256, 0, stream>>>(out, m, denom);

    gemm_wmma_kernel<<<N_NODES / 16, 128, 0, stream>>>(x, W, h);   // 50000/16 = 3125

    dots_kernel<<<(N_NODES * HEADS + 255) / 256, 256, 0, stream>>>(h, sa, da, s, dvec);

    int eh = TOT_E * HEADS;
    edge_max_kernel<<<(eh + 255) / 256, 256, 0, stream>>>(ei, s, dvec, m, e);
    edge_sum_kernel<<<(eh + 255) / 256, 256, 0, stream>>>(ei, m, denom, e);

    int ee = TOT_E * OUT_DIM;
    edge_agg_kernel<<<(ee + 255) / 256, 256, 0, stream>>>(ei, e, denom, h, out);
}